// AutoencoderP2CPDistance_4939212390978
// MI455X (gfx1250) — compile-verified
//
#include <hip/hip_runtime.h>
#include <hip/hip_bf16.h>
#include <math.h>

typedef __attribute__((ext_vector_type(2))) float v2f;
typedef __attribute__((ext_vector_type(8))) float v8f;

#define NPTS 2048            // points per batch (in_features/2)
#define NT   128             // 16x16 tiles per dimension (2048/16)
#define BLK  512             // 16 waves of 32
#define NWAVE (BLK / 32)
#define MROWS 4              // row tiles per wave per pass (B reuse factor)
#define BIGF 3.0e38f

// Hand-picked min ops: avoids clang's conservative canonicalize (v_max x,x)
// insertion around llvm.minnum for loop-carried values. All inputs here are
// canonical (WMMA outputs / mins thereof), so plain min_num is exact.
__device__ __forceinline__ float min3f(float a, float b, float c) {
    float d;
    asm("v_min3_num_f32 %0, %1, %2, %3" : "=v"(d) : "v"(a), "v"(b), "v"(c));
    return d;
}
__device__ __forceinline__ float min2f(float a, float b) {
    float d;
    asm("v_min_num_f32 %0, %1, %2" : "=v"(d) : "v"(a), "v"(b));
    return d;
}

// One workgroup per batch element. Computes 0.5*(mean_u min_v d + mean_v min_u d).
// dist^2 tile via a single V_WMMA_F32_16X16X4_F32:
//   A row M  = (-2ux, -2uy, ||u||^2, 1)     (K=4)
//   B col N  = ( vx ,  vy ,   1   , ||v||^2)
//   D[M][N]  = ||u||^2 + ||v||^2 - 2 u.v = dist^2
__global__ __launch_bounds__(BLK)
void p2cp_batch_kernel(const float* __restrict__ outs,
                       const float* __restrict__ tgts,
                       float* __restrict__ ws) {
    __shared__ v2f  sA[NPTS];        // {-2ux, -2uy}                       16KB
    __shared__ v2f  sB[2 * NPTS];    // [n]={vx,vy} ; [NPTS+n]={1,||v||^2}  32KB
    __shared__ int  sColMin[NPTS];   // d^2 bit patterns                    8KB
    __shared__ float sRowSum;
    __shared__ float sColSum;

    const int b    = blockIdx.x;
    const int tid  = threadIdx.x;
    const int lane = tid & 31;
    const int wave = tid >> 5;
    const int half = lane >> 4;     // 0: lanes 0-15, 1: lanes 16-31
    const int l15  = lane & 15;

    // Stage point clouds (vectorized B128 loads). x[b,n], x[b,2048+n] = coords.
    const float4* ob4 = (const float4*)(outs + (size_t)b * (2 * NPTS));
    const float4* tb4 = (const float4*)(tgts + (size_t)b * (2 * NPTS));
    for (int j = tid; j < NPTS / 4; j += BLK) {
        float4 ux = ob4[j];
        float4 uy = ob4[NPTS / 4 + j];
        float4 vx = tb4[j];
        float4 vy = tb4[NPTS / 4 + j];
        const float* uxp = (const float*)&ux;
        const float* uyp = (const float*)&uy;
        const float* vxp = (const float*)&vx;
        const float* vyp = (const float*)&vy;
        #pragma unroll
        for (int k = 0; k < 4; ++k) {
            int n = 4 * j + k;
            v2f ap; ap.x = -2.0f * uxp[k]; ap.y = -2.0f * uyp[k];
            sA[n] = ap;
            v2f b0; b0.x = vxp[k]; b0.y = vyp[k];
            sB[n] = b0;
            v2f b1; b1.x = 1.0f; b1.y = vxp[k] * vxp[k] + vyp[k] * vyp[k];
            sB[NPTS + n] = b1;
            sColMin[n] = 0x7F800000;   // +inf (positive int)
        }
    }
    if (tid == 0) { sRowSum = 0.0f; sColSum = 0.0f; }
    __syncthreads();

    const int bbase = half ? NPTS : 0;   // per-lane B-pack base, fixed
    float rowsum_local = 0.0f;

    // Two passes; each wave owns MROWS contiguous row tiles per pass.
    for (int ug = 0; ug < NT; ug += NWAVE * MROWS) {
        v2f a[MROWS];
        v8f rmin[MROWS];
        #pragma unroll
        for (int r = 0; r < MROWS; ++r) {
            const int m = (ug + wave * MROWS + r) * 16 + l15;
            v2f p = sA[m];
            float ar = 0.25f * (p.x * p.x + p.y * p.y);   // ||u||^2
            a[r].x = half ? ar   : p.x;
            a[r].y = half ? 1.0f : p.y;
            #pragma unroll
            for (int i = 0; i < 8; ++i) rmin[r][i] = BIGF;
        }

        // Double-prefetched B fragments (two col tiles per iteration).
        v2f bf0 = sB[bbase + l15];
        v2f bf1 = sB[bbase + 16 + l15];
        for (int vt = 0; vt < NT; vt += 2) {
            const v2f cur0 = bf0, cur1 = bf1;
            const int nx = (vt + 2) & (NT - 1);             // wrap: no branch
            bf0 = sB[bbase + nx * 16 + l15];
            bf1 = sB[bbase + (nx + 1) * 16 + l15];

            float cm0, cm1;
            #pragma unroll
            for (int r = 0; r < MROWS; ++r) {
                v8f c0 = {}, c1 = {};
                c0 = __builtin_amdgcn_wmma_f32_16x16x4_f32(
                        false, a[r], false, cur0, (short)0, c0, false, false);
                c1 = __builtin_amdgcn_wmma_f32_16x16x4_f32(
                        false, a[r], false, cur1, (short)0, c1, false, false);
                // row-min: one min3 folds both col tiles
                #pragma unroll
                for (int i = 0; i < 8; ++i)
                    rmin[r][i] = min3f(rmin[r][i], c0[i], c1[i]);
                // col-min trees (separate N tiles -> separate accumulators)
                float t0 = min3f(c0[0], c0[1], c0[2]);
                float t1 = min3f(c0[3], c0[4], c0[5]);
                float t2 = min3f(c0[6], c0[7], t0);
                float s0 = min3f(c1[0], c1[1], c1[2]);
                float s1 = min3f(c1[3], c1[4], c1[5]);
                float s2 = min3f(c1[6], c1[7], s0);
                if (r == 0) { cm0 = min2f(t1, t2);       cm1 = min2f(s1, s2); }
                else        { cm0 = min3f(cm0, t1, t2);  cm1 = min3f(cm1, s1, s2); }
            }
            // one atomic per 4 row tiles per col tile; signed-int min == float
            // min for >=0, -eps artifacts clamp to 0 below (correct limit).
            atomicMin(&sColMin[vt * 16 + l15],       (int)__float_as_uint(cm0));
            atomicMin(&sColMin[(vt + 1) * 16 + l15], (int)__float_as_uint(cm1));
        }

        // Row-min: butterfly across the 16 lanes of each half (masks<16 stay
        // inside a half). VGPR i of half h in tile r is row M = ut_r*16 + 8h + i.
        #pragma unroll
        for (int r = 0; r < MROWS; ++r) {
            #pragma unroll
            for (int i = 0; i < 8; ++i) {
                float v = rmin[r][i];
                v = min2f(v, __shfl_xor(v, 1));
                v = min2f(v, __shfl_xor(v, 2));
                v = min2f(v, __shfl_xor(v, 4));
                v = min2f(v, __shfl_xor(v, 8));
                rowsum_local += (l15 == 0) ? sqrtf(fmaxf(v, 0.0f)) : 0.0f;
            }
        }
    }

    atomicAdd(&sRowSum, rowsum_local);
    __syncthreads();

    float colsum_local = 0.0f;
    for (int i = tid; i < NPTS; i += BLK)
        colsum_local += sqrtf(fmaxf(__uint_as_float((unsigned)sColMin[i]), 0.0f));
    atomicAdd(&sColSum, colsum_local);
    __syncthreads();

    if (tid == 0)
        ws[b] = 0.5f * (sRowSum + sColSum) * (1.0f / (float)NPTS);
}

// Reduce the per-batch values to the scalar mean. Single wave32.
__global__ void p2cp_finalize_kernel(const float* __restrict__ ws,
                                     float* __restrict__ out, int bs) {
    int l = threadIdx.x;
    float v = (l < bs) ? ws[l] : 0.0f;
    v += __shfl_xor(v, 16);
    v += __shfl_xor(v, 8);
    v += __shfl_xor(v, 4);
    v += __shfl_xor(v, 2);
    v += __shfl_xor(v, 1);
    if (l == 0) out[0] = v / (float)bs;
}

extern "C" void kernel_launch(void* const* d_in, const int* in_sizes, int n_in,
                              void* d_out, int out_size, void* d_ws, size_t ws_size,
                              hipStream_t stream) {
    const float* outs = (const float*)d_in[0];
    const float* tgts = (const float*)d_in[1];
    float* out = (float*)d_out;
    float* ws  = (float*)d_ws;

    const int bs = in_sizes[0] / (2 * NPTS);   // 32 for the reference shapes

    p2cp_batch_kernel<<<bs, BLK, 0, stream>>>(outs, tgts, ws);
    p2cp_finalize_kernel<<<1, 32, 0, stream>>>(ws, out, bs);
}